// New_CALayer_60842506715326
// MI455X (gfx1250) — compile-verified
//
#include <hip/hip_runtime.h>
#include <math.h>

typedef __attribute__((ext_vector_type(2))) float v2f;
typedef __attribute__((ext_vector_type(8))) float v8f;

#define BD     2
#define LD     4096
#define DD     256
#define NHEADS 8
#define DHD    32
#define HSP    64
#define WSP    64
#define NROW   (BD * LD)   // 8192 rows

// ---- optional CDNA5 async global->LDS copy path (graceful fallback) -------
#if defined(__has_builtin)
#  if __has_builtin(__builtin_amdgcn_global_load_async_to_lds_b128) && \
      __has_builtin(__builtin_amdgcn_s_wait_asynccnt)
#    define USE_ASYNC_LDS 1
#  endif
#endif

#ifdef USE_ASYNC_LDS
typedef int v4i_async __attribute__((vector_size(16)));

__device__ __forceinline__ void async_cp16(const float* g, float* l) {
  __builtin_amdgcn_global_load_async_to_lds_b128(
      (__attribute__((address_space(1))) v4i_async*)(v4i_async*)const_cast<float*>(g),
      (__attribute__((address_space(3))) v4i_async*)(v4i_async*)l,
      0, 0);
}
#endif

// ---------------------------------------------------------------------------
// elementwise add: q = tgt + pos
// ---------------------------------------------------------------------------
__global__ void add_kernel(const float* __restrict__ a, const float* __restrict__ b,
                           float* __restrict__ o, int n) {
  int i = blockIdx.x * blockDim.x + threadIdx.x;
  if (i < n) o[i] = a[i] + b[i];
}

// ---------------------------------------------------------------------------
// C[M,N] = A[M,K] @ W[K,N] + bias[N]   via V_WMMA_F32_16X16X4_F32
// One 16x16 C tile per wave, 4 waves per block.
// f32 A 16x4 layout: lane = M%16, koff = (lane<16)?0:2, VGPR v -> K = koff+v
// f32 B 4x16 layout: col = lane&15, same koff scheme
// C layout: col = lane&15, VGPR i -> row i (+8 for lanes>=16)
// Prefetches are speculative on gfx1250 (silently dropped on bad translation)
// so they are issued branch-free, once per 16-k outer step.
// ---------------------------------------------------------------------------
__global__ __launch_bounds__(128)
void gemm_bias_wmma(const float* __restrict__ A, const float* __restrict__ W,
                    const float* __restrict__ bias, float* __restrict__ C,
                    int M, int N, int K) {
  int lane   = threadIdx.x & 31;
  int wave   = threadIdx.x >> 5;
  int tilesN = N >> 4;
  int tile   = blockIdx.x * 4 + wave;
  int total  = (M >> 4) * tilesN;
  if (tile >= total) return;               // wave-uniform
  int tm  = tile / tilesN;
  int tn  = tile - tm * tilesN;
  int lh  = (lane < 16) ? 0 : 2;
  int l15 = lane & 15;
  int col = tn * 16 + l15;
  const float* Arow = A + (size_t)(tm * 16 + l15) * K;

  v8f acc = {};
  for (int k0 = 0; k0 < K; k0 += 16) {
    // branch-free speculative prefetch: W 16 rows ahead, A 64 floats ahead
    __builtin_prefetch(&W[(size_t)(k0 + 16 + lh) * N + col], 0, 3);
    __builtin_prefetch(Arow + k0 + 64, 0, 3);
#pragma unroll
    for (int ki = 0; ki < 16; ki += 4) {
      int k = k0 + ki;
      float2 av = *(const float2*)(Arow + k + lh);    // even k -> 8B aligned
      v2f a; a.x = av.x; a.y = av.y;
      v2f b;
      b.x = W[(size_t)(k + lh)     * N + col];
      b.y = W[(size_t)(k + lh + 1) * N + col];
      acc = __builtin_amdgcn_wmma_f32_16x16x4_f32(false, a, false, b,
                                                  (short)0, acc, false, false);
    }
  }
  float bs = bias[col];
  int rbase = tm * 16 + ((lane < 16) ? 0 : 8);
#pragma unroll
  for (int i = 0; i < 8; ++i)
    C[(size_t)(rbase + i) * N + col] = acc[i] + bs;
}

// ---------------------------------------------------------------------------
// Deformable sampling: per-thread (b, lq, head, dh)
// loc = ref + off/64 ; grid_sample(align_corners=False, zeros) ; 4-pt softmax
// ---------------------------------------------------------------------------
__global__ __launch_bounds__(256)
void deform_sample(const float* __restrict__ value, const float* __restrict__ ref,
                   const float* __restrict__ off,   const float* __restrict__ aw,
                   float* __restrict__ samp) {
  int t = blockIdx.x * blockDim.x + threadIdx.x;
  if (t >= NROW * NHEADS * DHD) return;
  int dh  = t & 31;
  int h   = (t >> 5) & 7;
  int row = t >> 8;            // b*L + lq
  int b   = row >> 12;         // row / 4096

  float rx = ref[row * 2 + 0];
  float ry = ref[row * 2 + 1];

  float w[4];
  float mx = -1e30f;
#pragma unroll
  for (int p = 0; p < 4; ++p) { w[p] = aw[(size_t)row * 32 + h * 4 + p]; mx = fmaxf(mx, w[p]); }
  float ssum = 0.f;
#pragma unroll
  for (int p = 0; p < 4; ++p) { w[p] = __expf(w[p] - mx); ssum += w[p]; }
  float inv = 1.0f / ssum;

  const float* vbase = value + (size_t)b * LD * DD + h * DHD + dh;
  float out = 0.f;
#pragma unroll
  for (int p = 0; p < 4; ++p) {
    float ox = off[(size_t)row * 64 + h * 8 + p * 2 + 0];
    float oy = off[(size_t)row * 64 + h * 8 + p * 2 + 1];
    float x = (rx + ox * (1.0f / WSP)) * WSP - 0.5f;
    float y = (ry + oy * (1.0f / HSP)) * HSP - 0.5f;
    float x0f = floorf(x), y0f = floorf(y);
    float wx = x - x0f, wy = y - y0f;
    int x0 = (int)x0f, y0 = (int)y0f;

    float g00 = 0.f, g01 = 0.f, g10 = 0.f, g11 = 0.f;
    bool xi0 = (x0 >= 0) & (x0 < WSP);
    bool xi1 = (x0 + 1 >= 0) & (x0 + 1 < WSP);
    bool yi0 = (y0 >= 0) & (y0 < HSP);
    bool yi1 = (y0 + 1 >= 0) & (y0 + 1 < HSP);
    if (yi0 & xi0) g00 = vbase[(size_t)(y0 * WSP + x0) * DD];
    if (yi0 & xi1) g01 = vbase[(size_t)(y0 * WSP + x0 + 1) * DD];
    if (yi1 & xi0) g10 = vbase[(size_t)((y0 + 1) * WSP + x0) * DD];
    if (yi1 & xi1) g11 = vbase[(size_t)((y0 + 1) * WSP + x0 + 1) * DD];

    float top = g00 * (1.f - wx) + g01 * wx;
    float bot = g10 * (1.f - wx) + g11 * wx;
    out += (w[p] * inv) * (top * (1.f - wy) + bot * wy);
  }
  samp[(size_t)row * DD + h * DHD + dh] = out;
}

// ---------------------------------------------------------------------------
// o = LayerNorm(a + b) * g + beta ; one 256-wide row per 256-thread block
// ---------------------------------------------------------------------------
__global__ __launch_bounds__(256)
void add_ln(const float* __restrict__ a, const float* __restrict__ b,
            const float* __restrict__ g, const float* __restrict__ beta,
            float* __restrict__ o) {
  __shared__ float red[256];
  int row = blockIdx.x;
  int i   = threadIdx.x;
  float x = a[(size_t)row * DD + i] + b[(size_t)row * DD + i];

  red[i] = x; __syncthreads();
  for (int s = 128; s > 0; s >>= 1) { if (i < s) red[i] += red[i + s]; __syncthreads(); }
  float mu = red[0] * (1.0f / DD);
  __syncthreads();
  float d = x - mu;
  red[i] = d * d; __syncthreads();
  for (int s = 128; s > 0; s >>= 1) { if (i < s) red[i] += red[i + s]; __syncthreads(); }
  float var = red[0] * (1.0f / DD);
  o[(size_t)row * DD + i] = d * rsqrtf(var + 1e-5f) * g[i] + beta[i];
}

// ---------------------------------------------------------------------------
// Flash attention: grid = (L/16/ATQT, B*HEADS); ATQT waves per block, each
// wave owns one 16-query tile; K/V 16x32 tiles are cooperatively staged in
// LDS (async global->LDS copies when available), shared by all waves.
// S tile 16x16 via 8 WMMA (K=32), online softmax via shfl_xor within 16-lane
// halves, P staged through per-wave LDS to re-layout C->A, PV via 8 WMMA.
// ---------------------------------------------------------------------------
#define ATQT 2   // query tiles (waves) per block

__global__ __launch_bounds__(32 * ATQT)
void attn_kernel(const float* __restrict__ Q, const float* __restrict__ Kc,
                 const float* __restrict__ V, float* __restrict__ O) {
  __shared__ float kx[16 * 32];          // K tile  [key][dim]
  __shared__ float vx[16 * 32];          // V tile  [key][dim]
  __shared__ float px[ATQT][16 * 16];    // per-wave P tile [m][n]
  int tid  = threadIdx.x;
  int lane = tid & 31;
  int wave = tid >> 5;
  int qt   = blockIdx.x * ATQT + wave;
  int bh   = blockIdx.y;
  int b    = bh >> 3, h = bh & 7;
  int lh   = (lane < 16) ? 0 : 2;
  int l15  = lane & 15;
  const float scale = 0.17677669529663687f;  // 1/sqrt(32)

  // cooperative staging map: 16 rows x 8 float4 = 128 float4 per tile,
  // 64 threads -> 2 float4 per thread per tile
  const float* kgp = Kc + (size_t)(b * LD) * DD + h * DHD;
  const float* vgp = V  + (size_t)(b * LD) * DD + h * DHD;
  int sr0 = tid >> 3,        sc0 = (tid & 7) * 4;        // idx = tid
  int sr1 = (tid + 64) >> 3, sc1 = ((tid + 64) & 7) * 4; // idx = tid+64

  v2f qa[8];
  {
    const float* qp = Q + (size_t)(b * LD + qt * 16 + l15) * DD + h * DHD;
#pragma unroll
    for (int kk = 0; kk < 8; ++kk) {
      float2 t = *(const float2*)(qp + kk * 4 + lh);
      qa[kk].x = t.x; qa[kk].y = t.y;
    }
  }

  v8f acc0 = {}, acc1 = {};
  float m[8], l[8];
#pragma unroll
  for (int i = 0; i < 8; ++i) { m[i] = -1e30f; l[i] = 0.f; }

  for (int kt = 0; kt < LD / 16; ++kt) {
    __syncthreads();   // protect LDS tiles from previous iteration readers
    {
      size_t g0 = (size_t)(kt * 16 + sr0) * DD + sc0;
      size_t g1 = (size_t)(kt * 16 + sr1) * DD + sc1;
#ifdef USE_ASYNC_LDS
      async_cp16(kgp + g0, &kx[sr0 * 32 + sc0]);
      async_cp16(kgp + g1, &kx[sr1 * 32 + sc1]);
      async_cp16(vgp + g0, &vx[sr0 * 32 + sc0]);
      async_cp16(vgp + g1, &vx[sr1 * 32 + sc1]);
      __builtin_amdgcn_s_wait_asynccnt(0);
#else
      *(float4*)&kx[sr0 * 32 + sc0] = *(const float4*)(kgp + g0);
      *(float4*)&kx[sr1 * 32 + sc1] = *(const float4*)(kgp + g1);
      *(float4*)&vx[sr0 * 32 + sc0] = *(const float4*)(vgp + g0);
      *(float4*)&vx[sr1 * 32 + sc1] = *(const float4*)(vgp + g1);
#endif
    }
    __syncthreads();   // staged tiles visible to all waves

    // ---- S = (Q Kt^T) * scale ----
    v8f s = {};
#pragma unroll
    for (int kk = 0; kk < 8; ++kk) {
      float2 t = *(const float2*)(&kx[l15 * 32 + kk * 4 + lh]);
      v2f bb; bb.x = t.x; bb.y = t.y;
      s = __builtin_amdgcn_wmma_f32_16x16x4_f32(false, qa[kk], false, bb,
                                                (short)0, s, false, false);
    }
    // ---- online softmax (rows striped: vgpr i -> row i (+8 upper half)) ----
    float p[8];
#pragma unroll
    for (int i = 0; i < 8; ++i) {
      float sv = s[i] * scale;
      float mx = sv;
      mx = fmaxf(mx, __shfl_xor(mx, 1));
      mx = fmaxf(mx, __shfl_xor(mx, 2));
      mx = fmaxf(mx, __shfl_xor(mx, 4));
      mx = fmaxf(mx, __shfl_xor(mx, 8));
      float mn = fmaxf(m[i], mx);
      float f  = __expf(m[i] - mn);
      float pv = __expf(sv - mn);
      float rs = pv;
      rs += __shfl_xor(rs, 1);
      rs += __shfl_xor(rs, 2);
      rs += __shfl_xor(rs, 4);
      rs += __shfl_xor(rs, 8);
      l[i] = l[i] * f + rs;
      m[i] = mn;
      acc0[i] *= f; acc1[i] *= f;
      p[i] = pv;
    }
    // ---- stage P (C layout) -> per-wave LDS [m][n] (in-order within wave)
    int rb = (lane < 16) ? 0 : 8;
#pragma unroll
    for (int i = 0; i < 8; ++i) px[wave][(rb + i) * 16 + l15] = p[i];
    // ---- acc += P @ V  (V from LDS tile) ----
#pragma unroll
    for (int kk = 0; kk < 4; ++kk) {
      float2 t = *(const float2*)(&px[wave][l15 * 16 + kk * 4 + lh]);
      v2f pa; pa.x = t.x; pa.y = t.y;
      const float* v0 = &vx[(kk * 4 + lh) * 32];
      const float* v1 = v0 + 32;
      v2f b0; b0.x = v0[l15];      b0.y = v1[l15];
      v2f b1; b1.x = v0[16 + l15]; b1.y = v1[16 + l15];
      acc0 = __builtin_amdgcn_wmma_f32_16x16x4_f32(false, pa, false, b0,
                                                   (short)0, acc0, false, false);
      acc1 = __builtin_amdgcn_wmma_f32_16x16x4_f32(false, pa, false, b1,
                                                   (short)0, acc1, false, false);
    }
  }
  int rbase = qt * 16 + ((lane < 16) ? 0 : 8);
#pragma unroll
  for (int i = 0; i < 8; ++i) {
    float invl = 1.0f / l[i];
    size_t base = (size_t)(b * LD + rbase + i) * DD + h * DHD;
    O[base + l15]      = acc0[i] * invl;
    O[base + 16 + l15] = acc1[i] * invl;
  }
}

// ---------------------------------------------------------------------------
// host-side driver
// ---------------------------------------------------------------------------
static inline void launch_gemm(const float* A, const float* W, const float* bias,
                               float* C, int M, int N, int K, hipStream_t s) {
  int tiles = (M / 16) * (N / 16);
  int blocks = (tiles + 3) / 4;
  gemm_bias_wmma<<<blocks, 128, 0, s>>>(A, W, bias, C, M, N, K);
}

extern "C" void kernel_launch(void* const* d_in, const int* in_sizes, int n_in,
                              void* d_out, int out_size, void* d_ws, size_t ws_size,
                              hipStream_t stream) {
  const float* tgt  = (const float*)d_in[0];
  const float* pos  = (const float*)d_in[1];
  const float* ref  = (const float*)d_in[2];
  // d_in[3] spatial_shapes, d_in[4] level_start_index: static (64,64 / 0)
  const float* Wv   = (const float*)d_in[5];   const float* bv   = (const float*)d_in[6];
  const float* Woff = (const float*)d_in[7];   const float* boff = (const float*)d_in[8];
  const float* Waw  = (const float*)d_in[9];   const float* baw  = (const float*)d_in[10];
  const float* Wo   = (const float*)d_in[11];  const float* bo   = (const float*)d_in[12];
  const float* g1   = (const float*)d_in[13];  const float* b1   = (const float*)d_in[14];
  const float* Wq   = (const float*)d_in[15];  const float* bq   = (const float*)d_in[16];
  const float* Wk   = (const float*)d_in[17];  const float* bk   = (const float*)d_in[18];
  const float* Wvc  = (const float*)d_in[19];  const float* bvc  = (const float*)d_in[20];
  const float* Wp   = (const float*)d_in[21];  const float* bp   = (const float*)d_in[22];

  float* ws = (float*)d_ws;
  const size_t NE = (size_t)NROW * DD;     // 2,097,152
  float* q    = ws;                        // [8192,256]
  float* val  = ws + NE;                   // [8192,256]
  float* off  = ws + 2 * NE;               // [8192, 64]
  float* aw   = off + (size_t)NROW * 64;   // [8192, 32]
  float* samp = aw  + (size_t)NROW * 32;   // [8192,256]
  float* t2   = samp + NE;                 // [8192,256]
  float* t4   = t2   + NE;                 // [8192,256]
  float* qc   = t4   + NE;                 // [8192,256]
  float* kc   = qc   + NE;                 // [8192,256]
  float* vc   = kc   + NE;                 // [8192,256]
  float* ao   = vc   + NE;                 // [8192,256]
  float* t5   = ao   + NE;                 // [8192,256]
  // total ~87 MB of d_ws

  // 1. q = tgt + pos
  add_kernel<<<(int)((NE + 255) / 256), 256, 0, stream>>>(tgt, pos, q, (int)NE);
  // 2. projections for deformable attention
  launch_gemm(tgt, Wv,   bv,   val, NROW, DD, DD, stream);
  launch_gemm(q,   Woff, boff, off, NROW, 64, DD, stream);
  launch_gemm(q,   Waw,  baw,  aw,  NROW, 32, DD, stream);
  // 3. bilinear sampling + point softmax
  {
    int nt = NROW * NHEADS * DHD;
    deform_sample<<<(nt + 255) / 256, 256, 0, stream>>>(val, ref, off, aw, samp);
  }
  // 4. output proj + residual LN -> tgt4
  launch_gemm(samp, Wo, bo, t2, NROW, DD, DD, stream);
  add_ln<<<NROW, 256, 0, stream>>>(tgt, t2, g1, b1, t4);
  // 5. cross-attention projections
  launch_gemm(t4,  Wq,  bq,  qc, NROW, DD, DD, stream);
  launch_gemm(tgt, Wk,  bk,  kc, NROW, DD, DD, stream);
  launch_gemm(tgt, Wvc, bvc, vc, NROW, DD, DD, stream);
  // 6. flash attention
  {
    dim3 grid(LD / 16 / ATQT, BD * NHEADS);
    attn_kernel<<<grid, 32 * ATQT, 0, stream>>>(qc, kc, vc, ao);
  }
  // 7. output proj + residual LN -> out
  launch_gemm(ao, Wp, bp, t5, NROW, DD, DD, stream);
  add_ln<<<NROW, 256, 0, stream>>>(t4, t5, g1, b1, (float*)d_out);
}